// SparseCSREncoder_64252710748372
// MI455X (gfx1250) — compile-verified
//
#include <hip/hip_runtime.h>

// SparseCSREncoder roundtrip == elementwise threshold mask:
//   out[i] = (|x[i]| > 0.5f) ? x[i] : 0.0f
// Pure streaming kernel: 256 MiB of HBM traffic, ~11.5 us floor at 23.3 TB/s.
// Strategy: B128 NT loads/stores, unconditional straight-line hot kernel
// (bounds handling hoisted to a separate remainder kernel so the hot loop has
// zero exec-mask manipulation), 4 loads in flight per thread, wave32 blocks.

#ifndef THRESH
#define THRESH 0.5f
#endif

// Native clang vector (NOT HIP_vector_type) so nontemporal builtins accept it.
typedef __attribute__((ext_vector_type(4))) float v4f;

static __device__ __forceinline__ float mask1f(float v) {
    return (__builtin_fabsf(v) > THRESH) ? v : 0.0f;
}

static __device__ __forceinline__ v4f mask4f(v4f v) {
    v4f r;
    r.x = mask1f(v.x);
    r.y = mask1f(v.y);
    r.z = mask1f(v.z);
    r.w = mask1f(v.w);
    return r;
}

// Hot kernel: NO bounds checks. Each block owns a contiguous tile of
// 1024 v4f (16 KiB in / 16 KiB out). 4 back-to-back global_load_b128 NT
// per thread (clause-able), then 4 global_store_b128 NT.
__global__ __launch_bounds__(256) void sparse_mask_full_kernel(
    const v4f* __restrict__ x, v4f* __restrict__ y) {
    const unsigned i0 = blockIdx.x * 1024u + threadIdx.x;

    v4f a = __builtin_nontemporal_load(&x[i0 + 0u]);
    v4f b = __builtin_nontemporal_load(&x[i0 + 256u]);
    v4f c = __builtin_nontemporal_load(&x[i0 + 512u]);
    v4f d = __builtin_nontemporal_load(&x[i0 + 768u]);

    a = mask4f(a);
    b = mask4f(b);
    c = mask4f(c);
    d = mask4f(d);

    __builtin_nontemporal_store(a, &y[i0 + 0u]);
    __builtin_nontemporal_store(b, &y[i0 + 256u]);
    __builtin_nontemporal_store(c, &y[i0 + 512u]);
    __builtin_nontemporal_store(d, &y[i0 + 768u]);
}

// Remainder kernel: scalar, bounds-checked (empty for the reference shape).
__global__ __launch_bounds__(256) void sparse_mask_rem_kernel(
    const float* __restrict__ x, float* __restrict__ y, int n) {
    int i = blockIdx.x * blockDim.x + threadIdx.x;
    if (i < n) {
        float v = __builtin_nontemporal_load(&x[i]);
        __builtin_nontemporal_store(mask1f(v), &y[i]);
    }
}

extern "C" void kernel_launch(void* const* d_in, const int* in_sizes, int n_in,
                              void* d_out, int out_size, void* d_ws, size_t ws_size,
                              hipStream_t stream) {
    (void)n_in; (void)out_size; (void)d_ws; (void)ws_size;

    const float* x = (const float*)d_in[0];
    float*       y = (float*)d_out;
    const int    n = in_sizes[0];            // 32*1024*1024 = 33,554,432

    // Whole tiles of 4096 floats (1024 v4f) per block.
    const int full_blocks = n / 4096;        // 8192 for the reference size
    if (full_blocks > 0) {
        sparse_mask_full_kernel<<<full_blocks, 256, 0, stream>>>(
            (const v4f*)x, (v4f*)y);
    }

    // Leftover elements (none for the reference size).
    const long done = (long)full_blocks * 4096;
    const int  rem  = (int)(n - done);
    if (rem > 0) {
        const int rblocks = (rem + 255) / 256;
        sparse_mask_rem_kernel<<<rblocks, 256, 0, stream>>>(
            x + done, y + done, rem);
    }
}